// TorsionPositionTransformer_38465727103612
// MI455X (gfx1250) — compile-verified
//
#include <hip/hip_runtime.h>

// ---------------------------------------------------------------------------
// Torsion -> frames -> atom37, MI455X (gfx1250, wave32).
// Bandwidth-bound (~79MB traffic, ~1.7 FLOP/B): optimize data movement, not math.
// WMMA is not applicable: per-atom 3x3 rotations are data-dependent gathers,
// no shared GEMM structure; the win is LDS store-staging + async table loads.
// ---------------------------------------------------------------------------

#define NT 256                       // 8 wave32 waves per workgroup

#define N_AA      21
#define DF_FLOATS (N_AA * 8 * 16)    // 2688 floats  (10752 B)
#define GI_INTS   (N_AA * 37)        // 777 ints     (3108 B)
#define LP_FLOATS (N_AA * 37 * 3)    // 2331 floats  (9324 B)
#define FR_STRIDE 97                 // 8 frames * 12 floats + 1 pad (odd mod 64 -> no bank conflicts)
#define STG_FLOATS (NT * 111)        // 37 atoms * 3 floats per residue

typedef float vfloat4 __attribute__((ext_vector_type(4)));

// ---- CDNA5 async global->LDS path (ASYNCcnt), guarded so we always compile ----
#if defined(__HIP_DEVICE_COMPILE__) && defined(__gfx1250__) && \
    __has_builtin(__builtin_amdgcn_global_load_async_to_lds_b32) && \
    __has_builtin(__builtin_amdgcn_s_wait_asynccnt)
#define USE_ASYNC_LDS 1
#endif

#ifdef USE_ASYNC_LDS
typedef __attribute__((address_space(1))) int gas_int;
typedef __attribute__((address_space(3))) int las_int;
__device__ __forceinline__ void async_cp_b32(const int* g, int* l) {
    __builtin_amdgcn_global_load_async_to_lds_b32((gas_int*)g, (las_int*)l, 0, 0);
}
#endif

__device__ __forceinline__ void load_table_dwords(const int* __restrict__ g,
                                                  int* __restrict__ l,
                                                  int n, int tid) {
#ifdef USE_ASYNC_LDS
    for (int i = tid; i < n; i += NT) async_cp_b32(g + i, l + i);
#else
    for (int i = tid; i < n; i += NT) l[i] = g[i];
#endif
}

__global__ __launch_bounds__(NT) void torsion_atom37_kernel(
    const int*   __restrict__ aatype,          // (N)
    const float* __restrict__ sin_cos,         // (N,7,2)
    const float* __restrict__ default_frames,  // (21,8,4,4)
    const int*   __restrict__ group_idx,       // (21,37)
    const float* __restrict__ lit_positions,   // (21,37,3)
    float*       __restrict__ out_atoms,       // (N,37,3)
    float*       __restrict__ out_t,           // (N,8,3)
    int nres)
{
    __shared__ float s_df[DF_FLOATS];
    __shared__ int   s_gi[GI_INTS];
    __shared__ float s_lp[LP_FLOATS];
    __shared__ float s_fr[NT * FR_STRIDE];     // per-residue 8 frames (R[9],T[3])
    __shared__ float s_stg[STG_FLOATS];        // atom staging for coalesced stores

    const int tid        = threadIdx.x;
    const int block_base = blockIdx.x * NT;
    const int n          = block_base + tid;
    int nvalid = nres - block_base;
    if (nvalid > NT) nvalid = NT;

    // ---- Phase 0: tables -> LDS (async DMA path on CDNA5) ----
    load_table_dwords((const int*)default_frames, (int*)s_df, DF_FLOATS, tid);
    load_table_dwords(group_idx,                  s_gi,       GI_INTS,   tid);
    load_table_dwords((const int*)lit_positions,  (int*)s_lp, LP_FLOATS, tid);
#ifdef USE_ASYNC_LDS
    __builtin_amdgcn_s_wait_asynccnt(0);
#endif
    __syncthreads();

    // ---- Phase 1: build 8 rigid frames per residue, store to LDS ----
    int aa = 0;
    if (tid < nvalid) {
        aa = aatype[n];

        float sc[14];
        const float2* scp = (const float2*)(sin_cos + (size_t)n * 14);  // 8B aligned
        #pragma unroll
        for (int i = 0; i < 7; ++i) { float2 v = scp[i]; sc[2*i] = v.x; sc[2*i+1] = v.y; }

        float Rc[9] = {}; float Tc[3] = {};   // chained chi frame
        float* myfr = &s_fr[tid * FR_STRIDE];

        #pragma unroll
        for (int k = 0; k < 8; ++k) {
            const float* df = &s_df[(aa * 8 + k) * 16];
            float a00=df[0], a01=df[1], a02=df[2],  t0=df[3];
            float a10=df[4], a11=df[5], a12=df[6],  t1=df[7];
            float a20=df[8], a21=df[9], a22=df[10], t2=df[11];
            float sn = (k == 0) ? 0.f : sc[2*(k-1)];
            float cs = (k == 0) ? 1.f : sc[2*(k-1)+1];

            // R = drot @ rotx(angle): col1' = c*col1 + s*col2 ; col2' = c*col2 - s*col1
            float R[9], T[3];
            R[0]=a00; R[1]=fmaf(cs,a01, sn*a02); R[2]=fmaf(cs,a02,-sn*a01);
            R[3]=a10; R[4]=fmaf(cs,a11, sn*a12); R[5]=fmaf(cs,a12,-sn*a11);
            R[6]=a20; R[7]=fmaf(cs,a21, sn*a22); R[8]=fmaf(cs,a22,-sn*a21);
            T[0]=t0;  T[1]=t1;  T[2]=t2;

            if (k >= 5) {                      // chain chi2..chi4
                float Rn[9], Tn[3];
                #pragma unroll
                for (int i = 0; i < 3; ++i) {
                    #pragma unroll
                    for (int j = 0; j < 3; ++j)
                        Rn[i*3+j] = fmaf(Rc[i*3+0], R[j],
                                    fmaf(Rc[i*3+1], R[3+j], Rc[i*3+2]*R[6+j]));
                    Tn[i] = fmaf(Rc[i*3+0], T[0],
                            fmaf(Rc[i*3+1], T[1],
                            fmaf(Rc[i*3+2], T[2], Tc[i])));
                }
                #pragma unroll
                for (int q = 0; q < 9; ++q) { R[q] = Rn[q]; Rc[q] = Rn[q]; }
                #pragma unroll
                for (int q = 0; q < 3; ++q) { T[q] = Tn[q]; Tc[q] = Tn[q]; }
            } else if (k == 4) {               // chi1 seeds the chain
                #pragma unroll
                for (int q = 0; q < 9; ++q) Rc[q] = R[q];
                #pragma unroll
                for (int q = 0; q < 3; ++q) Tc[q] = T[q];
            }

            float* f = myfr + k * 12;
            #pragma unroll
            for (int q = 0; q < 9; ++q) f[q] = R[q];
            #pragma unroll
            for (int q = 0; q < 3; ++q) f[9+q] = T[q];
        }
    }
    __syncthreads();

    // ---- Phase 2a: 37 atoms per residue -> staging LDS ----
    if (tid < nvalid) {
        const float* myfr = &s_fr[tid * FR_STRIDE];
        float* stg = &s_stg[tid * 111];
        for (int a = 0; a < 37; ++a) {
            int gsel = s_gi[aa * 37 + a];
            const float* f  = myfr + gsel * 12;          // dynamic index -> LDS, not scratch
            const float* lp = &s_lp[(aa * 37 + a) * 3];
            float lp0 = lp[0], lp1 = lp[1], lp2 = lp[2];
            // out_i = sum_j R[j][i]*lp[j] + T[i]   (reference applies R^T)
            stg[a*3+0] = fmaf(f[0], lp0, fmaf(f[3], lp1, fmaf(f[6], lp2, f[9])));
            stg[a*3+1] = fmaf(f[1], lp0, fmaf(f[4], lp1, fmaf(f[7], lp2, f[10])));
            stg[a*3+2] = fmaf(f[2], lp0, fmaf(f[5], lp1, fmaf(f[8], lp2, f[11])));
        }
    }

    // ---- Phase 2b: coalesced write of t outputs from frame LDS ----
    {
        size_t tb = (size_t)block_base * 24;
        int total = nvalid * 24;
        for (int i = tid; i < total; i += NT) {
            int r   = i / 24;
            int rem = i - r * 24;
            int k   = rem / 3;
            int j   = rem - k * 3;
            out_t[tb + i] = s_fr[r * FR_STRIDE + k * 12 + 9 + j];
        }
    }
    __syncthreads();

    // ---- Phase 3: coalesced 128-bit streaming stores of atom outputs ----
    {
        size_t ab = (size_t)block_base * 111;            // *4B = multiple of 16
        if (nvalid == NT) {
            const vfloat4* src = (const vfloat4*)s_stg;
            vfloat4* dst = (vfloat4*)(out_atoms + ab);
            for (int i = tid; i < (NT * 111) / 4; i += NT)
                __builtin_nontemporal_store(src[i], &dst[i]);
        } else {
            int total = nvalid * 111;
            for (int i = tid; i < total; i += NT) out_atoms[ab + i] = s_stg[i];
        }
    }
}

extern "C" void kernel_launch(void* const* d_in, const int* in_sizes, int n_in,
                              void* d_out, int out_size, void* d_ws, size_t ws_size,
                              hipStream_t stream) {
    const int*   aatype         = (const int*)  d_in[0];
    const float* sin_cos        = (const float*)d_in[1];
    const float* default_frames = (const float*)d_in[2];
    const int*   group_idx      = (const int*)  d_in[3];
    const float* lit_positions  = (const float*)d_in[4];

    const int nres = in_sizes[0];                 // B*L = 131072
    float* out_atoms = (float*)d_out;             // (N,37,3)
    float* out_t     = out_atoms + (size_t)nres * 37 * 3;  // (N,8,3)

    int blocks = (nres + NT - 1) / NT;            // 512
    torsion_atom37_kernel<<<blocks, NT, 0, stream>>>(
        aatype, sin_cos, default_frames, group_idx, lit_positions,
        out_atoms, out_t, nres);
}